// MaxMarginCriterion_51376398794825
// MI455X (gfx1250) — compile-verified
//
#include <hip/hip_runtime.h>
#include <hip/hip_bf16.h>

typedef __attribute__((ext_vector_type(2))) float v2f;
typedef __attribute__((ext_vector_type(8))) float v8f;

#define MARGIN 0.1f
#define CHOICES 2048
#define BLOCK 256   // 8 waves of 32; each thread owns 8 columns

__global__ void __launch_bounds__(1) mm_zero_out(float* out) {
    out[0] = 0.0f;
}

// One block per row. Computes row hinge loss; either stores to rowloss[row]
// (two-pass reduction path) or atomically accumulates (scaled) into out.
__global__ void __launch_bounds__(BLOCK)
mm_hinge_kernel(const float* __restrict__ cossim,
                const long long* __restrict__ target,
                float* __restrict__ rowloss,   // may be null -> atomic path
                float* __restrict__ out,
                float inv_n) {
    __shared__ float s_corr;
    __shared__ float s_wave[BLOCK / 32];

    const int row = blockIdx.x;
    const int tid = threadIdx.x;

    // ---- Pass over target (int64, one-hot). Read as int4 = 2 elements. ----
    const int4* __restrict__ tgt4 = (const int4*)target;
    const long long rowbase4 = (long long)row * (CHOICES / 2);  // int4 units
    #pragma unroll
    for (int s = 0; s < 4; ++s) {
        int col = s * 512 + tid * 2;
        int4 t = tgt4[rowbase4 + s * 256 + tid];
        // little-endian int64: value (0/1) sits in the low dword (x / z)
        if (t.x == 1) s_corr = cossim[(long long)row * CHOICES + col];
        if (t.z == 1) s_corr = cossim[(long long)row * CHOICES + col + 1];
    }

    __syncthreads();
    const float corr = s_corr;

    // ---- Pass over cossim: two float4 per thread (coalesced b128). ----
    const float4* __restrict__ cs4 = (const float4*)cossim;
    const long long rowbaseF4 = (long long)row * (CHOICES / 4);
    float4 c0 = cs4[rowbaseF4 + tid];          // cols [tid*4, tid*4+4)
    float4 c1 = cs4[rowbaseF4 + 256 + tid];    // cols [1024 + tid*4, ...)

    const float bias = MARGIN - corr;
    float h0 = fmaxf(c0.x + bias, 0.0f);
    float h1 = fmaxf(c0.y + bias, 0.0f);
    float h2 = fmaxf(c0.z + bias, 0.0f);
    float h3 = fmaxf(c0.w + bias, 0.0f);
    float h4 = fmaxf(c1.x + bias, 0.0f);
    float h5 = fmaxf(c1.y + bias, 0.0f);
    float h6 = fmaxf(c1.z + bias, 0.0f);
    float h7 = fmaxf(c1.w + bias, 0.0f);

    // ---- Wave-level reduction through the matrix pipe. ----
    // D = A x B + C with B = all-ones 4x16: D[m][n] = sum_k A[m][k].
    // Four chained WMMAs consume all 8 hinge values per lane. Column N=0
    // lives on lanes 0 and 16 (VGPRs 0..7 = M 0..7 / 8..15), so those two
    // lanes end up holding the complete wave sum between them.
    v2f b; b.x = 1.0f; b.y = 1.0f;
    v8f acc = {};
    v2f a;
    a.x = h0; a.y = h1;
    acc = __builtin_amdgcn_wmma_f32_16x16x4_f32(false, a, false, b, (short)0, acc, false, false);
    a.x = h2; a.y = h3;
    acc = __builtin_amdgcn_wmma_f32_16x16x4_f32(false, a, false, b, (short)0, acc, false, false);
    a.x = h4; a.y = h5;
    acc = __builtin_amdgcn_wmma_f32_16x16x4_f32(false, a, false, b, (short)0, acc, false, false);
    a.x = h6; a.y = h7;
    acc = __builtin_amdgcn_wmma_f32_16x16x4_f32(false, a, false, b, (short)0, acc, false, false);

    float lanesum = acc[0] + acc[1] + acc[2] + acc[3] +
                    acc[4] + acc[5] + acc[6] + acc[7];

    // lane 0 fetches lane 16's column-0 partial; their sum is the wave total
    float other = __shfl(lanesum, 16, 32);
    int lane = tid & 31;
    if (lane == 0) s_wave[tid >> 5] = lanesum + other;

    __syncthreads();
    if (tid == 0) {
        float rs = 0.0f;
        #pragma unroll
        for (int w = 0; w < BLOCK / 32; ++w) rs += s_wave[w];
        rs -= MARGIN;   // remove the correct-column term
        if (rowloss) rowloss[row] = rs;             // deterministic path
        else atomicAdd(out, rs * inv_n);            // fallback
    }
}

// Single-block deterministic reduction of n row losses -> mean.
__global__ void __launch_bounds__(BLOCK)
mm_reduce_kernel(const float* __restrict__ rowloss, float* __restrict__ out,
                 int n, float inv_n) {
    __shared__ float s[BLOCK];
    const int tid = threadIdx.x;
    const float4* __restrict__ rl4 = (const float4*)rowloss;
    const int n4 = n / 4;   // n is a multiple of 1024 here

    float acc = 0.0f;
    for (int i = tid; i < n4; i += BLOCK) {
        float4 v = rl4[i];
        acc += (v.x + v.y) + (v.z + v.w);
    }
    s[tid] = acc;
    __syncthreads();
    #pragma unroll
    for (int step = BLOCK / 2; step > 0; step >>= 1) {
        if (tid < step) s[tid] += s[tid + step];
        __syncthreads();
    }
    if (tid == 0) out[0] = s[0] * inv_n;
}

extern "C" void kernel_launch(void* const* d_in, const int* in_sizes, int n_in,
                              void* d_out, int out_size, void* d_ws, size_t ws_size,
                              hipStream_t stream) {
    const float* cossim = (const float*)d_in[0];
    const long long* target = (const long long*)d_in[1];
    float* out = (float*)d_out;

    const int n = in_sizes[0] / CHOICES;   // 16384 rows
    const float inv_n = 1.0f / (float)n;

    if (ws_size >= (size_t)n * sizeof(float)) {
        float* rowloss = (float*)d_ws;
        mm_hinge_kernel<<<n, BLOCK, 0, stream>>>(cossim, target, rowloss, out, inv_n);
        mm_reduce_kernel<<<1, BLOCK, 0, stream>>>(rowloss, out, n, inv_n);
    } else {
        mm_zero_out<<<1, 1, 0, stream>>>(out);
        mm_hinge_kernel<<<n, BLOCK, 0, stream>>>(cossim, target, nullptr, out, inv_n);
    }
}